// GCNWrapper_53257594471012
// MI455X (gfx1250) — compile-verified
//
#include <hip/hip_runtime.h>

#define IN_DIM   128
#define HID_DIM  128
#define EDGE_DIM 16

typedef float v2f __attribute__((ext_vector_type(2)));
typedef float v4f __attribute__((ext_vector_type(4)));
typedef float v8f __attribute__((ext_vector_type(8)));

// ---------------------------------------------------------------------------
// K1: scatter raw edge_attr into per-node accumulator (linearity of edge MLP)
//     agg[col] += edge_attr[e]  (16 floats), degf[col] += 1
// edge_attr is streamed exactly once -> non-temporal loads (keep L2 for xw).
// ---------------------------------------------------------------------------
__global__ void __launch_bounds__(256)
scatter_attr_kernel(const float* __restrict__ edge_attr,
                    const long long* __restrict__ edge_index,
                    float* __restrict__ agg, float* __restrict__ degf, int E)
{
    const int tid = blockIdx.x * 256 + threadIdx.x;
    const int e = tid >> 2;
    if (e >= E) return;
    const int q   = (tid & 3) * 4;
    const int col = (int)edge_index[(size_t)E + e];
    const v4f v = __builtin_nontemporal_load((const v4f*)(edge_attr + (size_t)e * EDGE_DIM + q));
    float* dst = agg + (size_t)col * EDGE_DIM + q;
    atomicAdd(dst + 0, v.x);
    atomicAdd(dst + 1, v.y);
    atomicAdd(dst + 2, v.z);
    atomicAdd(dst + 3, v.w);
    if (q == 0) atomicAdd(degf + col, 1.0f);
}

// ---------------------------------------------------------------------------
// K2: dis[n] = rsqrt(deg_total) with deg_total = degf + 1 (self loop)
// ---------------------------------------------------------------------------
__global__ void __launch_bounds__(256)
dis_kernel(const float* __restrict__ degf, float* __restrict__ dis, int N)
{
    const int n = blockIdx.x * 256 + threadIdx.x;
    if (n < N) dis[n] = rsqrtf(degf[n] + 1.0f);
}

// ---------------------------------------------------------------------------
// K3: fused dense part. 256 threads = 8 waves; W_conv (64 KB) staged once per
// block into LDS via CDNA5 async global->LDS copies (ASYNCcnt), then each
// wave32 computes one 16-node tile with fp32 WMMA 16x16x4:
//   x'  = x + agg @ W_edge^T + degf * b_edge        (K = 16)
//   xw  = x' @ W_conv^T                             (K = 128, B from LDS)
//   out = dis^2 * xw (self-loop init), xw stored for the SpMM.
// Fragment layouts (ISA 7.12.2, wave32):
//   A (16x4 f32): lane<16 -> K={kb,kb+1}, lane>=16 -> K={kb+2,kb+3}
//   B (4x16):     VGPR v, K = v + 2*(lane>>4), N = lane&15
//   C/D (16x16):  VGPR v, M = v + 8*(lane>>4), N = lane&15
// Dynamic LDS: 64 KB W_conv + 8 * 8 KB x' tiles = 128 KB (2 blocks / WGP).
// ---------------------------------------------------------------------------
#define K3_SMEM_BYTES ((IN_DIM * HID_DIM + 8 * 16 * IN_DIM) * sizeof(float))

__global__ void __launch_bounds__(256)
fused_gemm_kernel(const float* __restrict__ x,
                  const float* __restrict__ agg,
                  const float* __restrict__ degf,
                  const float* __restrict__ W_edge,   // [128][16]
                  const float* __restrict__ b_edge,   // [128]
                  const float* __restrict__ W_conv,   // [128][128]
                  float* __restrict__ xw,
                  float* __restrict__ out,
                  int N, int n_tiles)
{
    extern __shared__ char smem_raw[];
    float* sW    = (float*)smem_raw;                                   // [128][128]
    float* xpAll = (float*)(smem_raw + IN_DIM * HID_DIM * sizeof(float));

    const int wave = threadIdx.x >> 5;
    const int lane = threadIdx.x & 31;
    const int m    = lane & 15;             // A-row / B-col (N) index
    const int half = lane >> 4;
    float* xp = xpAll + wave * (16 * IN_DIM);

    // ---- async stage W_conv -> LDS (b128 per lane, 16 iterations) -------
    {
        const uint32_t lds0 = (uint32_t)(uintptr_t)sW;   // LDS byte offset
        #pragma unroll
        for (int it = 0; it < (IN_DIM * HID_DIM * 4) / (256 * 16); ++it) {
            const int      byte_off = (it * 256 + (int)threadIdx.x) * 16;
            const uint32_t ldsa     = lds0 + byte_off;
            const void*    gaddr    = (const char*)W_conv + byte_off;
            asm volatile("global_load_async_to_lds_b128 %0, %1, off"
                         :: "v"(ldsa), "v"(gaddr) : "memory");
        }
        asm volatile("s_wait_asynccnt 0" ::: "memory");
    }
    __syncthreads();   // all waves issued+completed their slices

    const int tile = blockIdx.x * 8 + wave;
    if (tile >= n_tiles) return;
    const int base = tile * 16;

    // ---- GEMM1: edge-MLP applied to aggregated attributes --------------
    const int am = min(base + m, N - 1);                 // clamped A row
    for (int j = 0; j < IN_DIM / 16; ++j) {
        v8f acc = {};
        #pragma unroll
        for (int kk = 0; kk < EDGE_DIM / 4; ++kk) {
            const int kb = kk * 4 + half * 2;
            v2f a = *(const v2f*)(agg + (size_t)am * EDGE_DIM + kb);
            v2f b = *(const v2f*)(W_edge + (size_t)(j * 16 + m) * EDGE_DIM + kb);
            acc = __builtin_amdgcn_wmma_f32_16x16x4_f32(
                false, a, false, b, (short)0, acc, false, false);
        }
        #pragma unroll
        for (int v = 0; v < 8; ++v) {
            const int r    = v + half * 8;
            const int node = min(base + r, N - 1);
            const int col  = j * 16 + m;
            const float xv = x[(size_t)node * IN_DIM + col];
            const float dv = degf[node];
            xp[r * IN_DIM + col] = xv + acc[v] + dv * b_edge[col];
        }
    }

    // DS ops are in-order within a wave; make the cross-lane RAW explicit.
    asm volatile("s_wait_dscnt 0" ::: "memory");

    // ---- GEMM2: xw = x' @ W_conv^T (A from LDS xp, B from LDS sW) ------
    for (int j = 0; j < HID_DIM / 16; ++j) {
        v8f acc = {};
        #pragma unroll
        for (int kk = 0; kk < IN_DIM / 4; ++kk) {
            const int kb = kk * 4 + half * 2;
            v2f a = *(const v2f*)(xp + m * IN_DIM + kb);
            v2f b = *(const v2f*)(sW + (j * 16 + m) * IN_DIM + kb);
            acc = __builtin_amdgcn_wmma_f32_16x16x4_f32(
                false, a, false, b, (short)0, acc, false, false);
        }
        #pragma unroll
        for (int v = 0; v < 8; ++v) {
            const int r    = v + half * 8;
            const int node = base + r;
            if (node < N) {
                const int col   = j * 16 + m;
                const float val = acc[v];
                xw[(size_t)node * HID_DIM + col] = val;
                // dis^2 = 1/deg_total : self-loop message initializes out
                out[(size_t)node * HID_DIM + col] = val / (degf[node] + 1.0f);
            }
        }
    }
}

// ---------------------------------------------------------------------------
// K4: SpMM propagation — one wave32 per edge, float4 per lane (128 floats).
//     out[col] += dis[row]*dis[col] * xw[row]   (xw stays L2-resident)
// ---------------------------------------------------------------------------
__global__ void __launch_bounds__(256)
propagate_kernel(const float* __restrict__ xw,
                 const long long* __restrict__ edge_index,
                 const float* __restrict__ dis,
                 float* __restrict__ out, int E)
{
    const int wave = threadIdx.x >> 5;
    const int lane = threadIdx.x & 31;
    const int e = blockIdx.x * 8 + wave;
    if (e >= E) return;
    const int row = (int)edge_index[e];
    const int col = (int)edge_index[(size_t)E + e];
    const float nrm = dis[row] * dis[col];
    const v4f v = *(const v4f*)(xw + (size_t)row * HID_DIM + lane * 4);
    float* dst = out + (size_t)col * HID_DIM + lane * 4;
    atomicAdd(dst + 0, nrm * v.x);
    atomicAdd(dst + 1, nrm * v.y);
    atomicAdd(dst + 2, nrm * v.z);
    atomicAdd(dst + 3, nrm * v.w);
}

// ---------------------------------------------------------------------------
// K5: out = relu(out + b_conv), in place
// ---------------------------------------------------------------------------
__global__ void __launch_bounds__(256)
finalize_kernel(float* __restrict__ out, const float* __restrict__ b_conv, int total)
{
    const int i = blockIdx.x * 256 + threadIdx.x;
    if (i < total) out[i] = fmaxf(out[i] + b_conv[i & (HID_DIM - 1)], 0.0f);
}

// ---------------------------------------------------------------------------
extern "C" void kernel_launch(void* const* d_in, const int* in_sizes, int n_in,
                              void* d_out, int out_size, void* d_ws, size_t ws_size,
                              hipStream_t stream)
{
    const float*     x          = (const float*)d_in[0];
    const float*     edge_attr  = (const float*)d_in[1];
    const float*     W_edge     = (const float*)d_in[2];
    const float*     b_edge     = (const float*)d_in[3];
    const float*     W_conv     = (const float*)d_in[4];
    const float*     b_conv     = (const float*)d_in[5];
    const long long* edge_index = (const long long*)d_in[6];
    float* out = (float*)d_out;

    const int N = in_sizes[0] / IN_DIM;
    const int E = in_sizes[1] / EDGE_DIM;

    // workspace layout: agg[N*16] | degf[N] | dis[N] | xw[N*128]
    float* agg  = (float*)d_ws;
    float* degf = agg + (size_t)N * EDGE_DIM;
    float* dis  = degf + N;
    float* xw   = dis + N;

    // zero the atomic accumulators (agg + degf are contiguous)
    hipMemsetAsync(agg, 0, sizeof(float) * (size_t)N * (EDGE_DIM + 1), stream);

    {   // K1: scatter edge attributes + degree
        const int th = E * 4;
        scatter_attr_kernel<<<(th + 255) / 256, 256, 0, stream>>>(
            edge_attr, edge_index, agg, degf, E);
    }
    // K2: dis = rsqrt(deg+1)
    dis_kernel<<<(N + 255) / 256, 256, 0, stream>>>(degf, dis, N);

    {   // K3: fused WMMA GEMMs with LDS-resident W_conv (128 KB dynamic LDS)
        const int n_tiles = (N + 15) / 16;
        fused_gemm_kernel<<<(n_tiles + 7) / 8, 256, K3_SMEM_BYTES, stream>>>(
            x, agg, degf, W_edge, b_edge, W_conv, xw, out, N, n_tiles);
    }
    // K4: SpMM propagation (one wave per edge)
    propagate_kernel<<<(E + 7) / 8, 256, 0, stream>>>(xw, edge_index, dis, out, E);

    // K5: bias + relu
    {
        const int total = N * HID_DIM;
        finalize_kernel<<<(total + 255) / 256, 256, 0, stream>>>(out, b_conv, total);
    }
}